// GCNNet1_5781025980782
// MI455X (gfx1250) — compile-verified
//
#include <hip/hip_runtime.h>

// ---- problem constants (match reference) ----
#define F_IN 128
#define HID  128
#define EMBD 64
#define NCLS 40

typedef float v2f __attribute__((ext_vector_type(2)));
typedef float v8f __attribute__((ext_vector_type(8)));

__device__ __forceinline__ void atomAddF(float* p, float v) {
  // hardware global_atomic_add_f32 path
  unsafeAtomicAdd(p, v);
}

// ---------------------------------------------------------------------------
// WMMA f32 GEMM:  C[nrows x NCOL] = A[nrows x K] * B[K x NCOL]
// Block = (16*MT) rows x NCOL cols. One wave per 16-col tile, MT row-tiles
// per wave so each B fragment is reused MT times (halves B-load overhead).
// Requires nrows % (16*MT) == 0  (N = 100000 = 3125 * 32 for MT=2).
// ---------------------------------------------------------------------------
template<int K, int NCT, int MT>
__global__ __launch_bounds__(32 * NCT) void gemm_wmma_f32(
    const float* __restrict__ A, const float* __restrict__ Bm,
    float* __restrict__ C) {
  constexpr int NCOL    = 16 * NCT;
  constexpr int THREADS = 32 * NCT;
  constexpr int ROWS    = 16 * MT;
  constexpr int LDA     = K + 4;          // pad to avoid LDS bank conflicts
  __shared__ float As[ROWS * LDA];

  const int tid        = threadIdx.x;
  const long long row0 = (long long)blockIdx.x * ROWS;

  // cooperative load of the ROWS x K A-tile into LDS (float4 granularity)
  for (int idx = tid * 4; idx < ROWS * K; idx += THREADS * 4) {
    const int r = idx / K;
    const int c = idx - r * K;
    const float4 v = *(const float4*)(A + (row0 + r) * K + c);
    float* dp = &As[r * LDA + c];
    dp[0] = v.x; dp[1] = v.y; dp[2] = v.z; dp[3] = v.w;
  }
  __syncthreads();

  const int wave = tid >> 5;
  const int lane = tid & 31;
  const int m    = lane & 15;              // A row within tile / B col within tile
  const int kq   = (lane >> 4) << 1;       // K sub-offset: 0 (lanes 0-15) or 2 (16-31)
  const int col  = (wave << 4) + m;        // global column

  v8f acc[MT];
#pragma unroll
  for (int t = 0; t < MT; ++t) acc[t] = (v8f){0.f,0.f,0.f,0.f,0.f,0.f,0.f,0.f};

#pragma unroll
  for (int k = 0; k < K; k += 4) {
    v2f b;
    b.x = Bm[(k + kq)     * NCOL + col];
    b.y = Bm[(k + kq + 1) * NCOL + col];
#pragma unroll
    for (int t = 0; t < MT; ++t) {
      v2f a;
      a.x = As[(m + 16 * t) * LDA + k + kq];
      a.y = As[(m + 16 * t) * LDA + k + kq + 1];
      acc[t] = __builtin_amdgcn_wmma_f32_16x16x4_f32(
          /*neg_a=*/false, a, /*neg_b=*/false, b,
          /*c_mod=*/(short)0, acc[t], /*reuse_a=*/false, /*reuse_b=*/false);
    }
  }

  // D layout: VGPR r -> row (r + 8*(lane>>4)), col = lane&15 within wave tile
#pragma unroll
  for (int t = 0; t < MT; ++t) {
    const long long mrow = row0 + 16 * t + ((lane >> 4) << 3);
#pragma unroll
    for (int r = 0; r < 8; ++r)
      C[(mrow + r) * NCOL + col] = acc[t][r];
  }
}

// ---------------------------------------------------------------------------
// Degree / normalization
// ---------------------------------------------------------------------------
__global__ void init_deg_kernel(float* __restrict__ deg, int n) {
  int i = blockIdx.x * blockDim.x + threadIdx.x;
  if (i < n) deg[i] = 1.0f;               // self loop
}

__global__ void count_deg_kernel(const long long* __restrict__ dst,
                                 float* __restrict__ deg, int e) {
  int i = blockIdx.x * blockDim.x + threadIdx.x;
  if (i < e) atomAddF(&deg[dst[i]], 1.0f);
}

__global__ void rsqrt_kernel(float* __restrict__ d, int n) {
  int i = blockIdx.x * blockDim.x + threadIdx.x;
  if (i < n) d[i] = rsqrtf(d[i]);         // deg >= 1 always
}

// ---------------------------------------------------------------------------
// Aggregation: agg = dinv_i^2 * h_i  (self loop)  then scatter-add over edges
// ---------------------------------------------------------------------------
template<int W>
__global__ void self_scale_kernel(const float* __restrict__ h,
                                  const float* __restrict__ dinv,
                                  float* __restrict__ agg, long long total) {
  long long t = (long long)blockIdx.x * blockDim.x + threadIdx.x;
  if (t < total) {
    long long i = t / W;
    float d = dinv[i];
    agg[t] = d * d * h[t];
  }
}

template<int W>
__global__ void edge_scatter_kernel(const long long* __restrict__ src,
                                    const long long* __restrict__ dst,
                                    const float* __restrict__ dinv,
                                    const float* __restrict__ h,
                                    float* __restrict__ agg, int nedges) {
  constexpr int TPE = W / 4;              // threads per edge (float4 each)
  long long t = (long long)blockIdx.x * blockDim.x + threadIdx.x;
  long long e = t / TPE;
  if (e >= nedges) return;
  int f = (int)(t - e * TPE) * 4;
  long long s = src[e], d = dst[e];
  float nrm = dinv[s] * dinv[d];
  const float4 v = *(const float4*)(h + s * W + f);
  float* o = agg + d * W + f;
  atomAddF(o + 0, nrm * v.x);
  atomAddF(o + 1, nrm * v.y);
  atomAddF(o + 2, nrm * v.z);
  atomAddF(o + 3, nrm * v.w);
}

template<int W, bool RELU>
__global__ void bias_act_kernel(const float* __restrict__ agg,
                                const float* __restrict__ b,
                                float* __restrict__ out, long long total) {
  long long t = (long long)blockIdx.x * blockDim.x + threadIdx.x;
  if (t < total) {
    float v = agg[t] + b[t & (W - 1)];
    out[t] = RELU ? fmaxf(v, 0.0f) : v;
  }
}

// ---------------------------------------------------------------------------
// Classifier + log_softmax: logits = emb @ Wl + bl ; out = log_softmax(logits)
// One thread per node; Wl (64x40) staged in LDS.
// ---------------------------------------------------------------------------
__global__ __launch_bounds__(128) void classifier_kernel(
    const float* __restrict__ emb, const float* __restrict__ Wl,
    const float* __restrict__ bl, float* __restrict__ out, int n) {
  __shared__ float Ws[EMBD * NCLS];
  __shared__ float bs[NCLS];
  for (int i = threadIdx.x; i < EMBD * NCLS; i += 128) Ws[i] = Wl[i];
  if (threadIdx.x < NCLS) bs[threadIdx.x] = bl[threadIdx.x];
  __syncthreads();

  int node = blockIdx.x * 128 + threadIdx.x;
  if (node >= n) return;

  float ev[EMBD];
  const float4* ep = (const float4*)(emb + (long long)node * EMBD);
#pragma unroll
  for (int j = 0; j < EMBD / 4; ++j) {
    float4 v = ep[j];
    ev[4 * j] = v.x; ev[4 * j + 1] = v.y; ev[4 * j + 2] = v.z; ev[4 * j + 3] = v.w;
  }

  float lg[NCLS];
  float mx = -3.4e38f;
#pragma unroll
  for (int o = 0; o < NCLS; ++o) {
    float acc = bs[o];
#pragma unroll
    for (int k = 0; k < EMBD; ++k) acc = fmaf(ev[k], Ws[k * NCLS + o], acc);
    lg[o] = acc;
    mx = fmaxf(mx, acc);
  }
  float s = 0.f;
#pragma unroll
  for (int o = 0; o < NCLS; ++o) s += __expf(lg[o] - mx);
  float lse = mx + __logf(s);
  float* op = out + (long long)node * NCLS;
#pragma unroll
  for (int o = 0; o < NCLS; ++o) op[o] = lg[o] - lse;
}

// ---------------------------------------------------------------------------
extern "C" void kernel_launch(void* const* d_in, const int* in_sizes, int n_in,
                              void* d_out, int out_size, void* d_ws, size_t ws_size,
                              hipStream_t stream) {
  const float*     x  = (const float*)d_in[0];
  const long long* ei = (const long long*)d_in[1];     // int64 edge_index (2,E)
  const float*     W1 = (const float*)d_in[2];
  const float*     b1 = (const float*)d_in[3];
  const float*     W2 = (const float*)d_in[4];
  const float*     b2 = (const float*)d_in[5];
  const float*     Wl = (const float*)d_in[6];
  const float*     bl = (const float*)d_in[7];

  const int n = in_sizes[0] / F_IN;        // 100000
  const int e = in_sizes[1] / 2;           // 1600000
  const long long* src = ei;
  const long long* dst = ei + e;

  // workspace layout: bufA (n*128) | bufB (n*128) | dinv (n)
  float* bufA = (float*)d_ws;
  float* bufB = bufA + (size_t)n * HID;
  float* dinv = bufB + (size_t)n * HID;

  float* logits = (float*)d_out;                    // n * 40
  float* emb    = logits + (size_t)n * NCLS;        // n * 64

  const int B = 256;
  // ---- degree / norm ----
  init_deg_kernel<<<(n + B - 1) / B, B, 0, stream>>>(dinv, n);
  count_deg_kernel<<<(e + B - 1) / B, B, 0, stream>>>(dst, dinv, e);
  rsqrt_kernel<<<(n + B - 1) / B, B, 0, stream>>>(dinv, n);

  // ---- conv1: h1 = x @ W1 ; agg ; relu(+b1) ----
  gemm_wmma_f32<F_IN, HID / 16, 2>
      <<<n / 32, 32 * (HID / 16), 0, stream>>>(x, W1, bufA);
  {
    long long tot = (long long)n * HID;
    self_scale_kernel<HID><<<(unsigned)((tot + B - 1) / B), B, 0, stream>>>(bufA, dinv, bufB, tot);
    long long et = (long long)e * (HID / 4);
    edge_scatter_kernel<HID><<<(unsigned)((et + B - 1) / B), B, 0, stream>>>(src, dst, dinv, bufA, bufB, e);
    bias_act_kernel<HID, true><<<(unsigned)((tot + B - 1) / B), B, 0, stream>>>(bufB, b1, bufB, tot);
  }

  // ---- conv2: h2 = h @ W2 ; agg ; +b2 -> embedding ----
  gemm_wmma_f32<HID, EMBD / 16, 2>
      <<<n / 32, 32 * (EMBD / 16), 0, stream>>>(bufB, W2, bufA);
  {
    long long tot = (long long)n * EMBD;
    self_scale_kernel<EMBD><<<(unsigned)((tot + B - 1) / B), B, 0, stream>>>(bufA, dinv, bufB, tot);
    long long et = (long long)e * (EMBD / 4);
    edge_scatter_kernel<EMBD><<<(unsigned)((et + B - 1) / B), B, 0, stream>>>(src, dst, dinv, bufA, bufB, e);
    bias_act_kernel<EMBD, false><<<(unsigned)((tot + B - 1) / B), B, 0, stream>>>(bufB, b2, emb, tot);
  }

  // ---- classifier + log_softmax ----
  classifier_kernel<<<(n + 127) / 128, 128, 0, stream>>>(emb, Wl, bl, logits, n);
}